// NMSLayer_42674795053126
// MI455X (gfx1250) — compile-verified
//
#include <hip/hip_runtime.h>
#include <hip/hip_bf16.h>
#include <math.h>

// ---------------------------------------------------------------------------
// Fused "normal angle" kernel for MI455X (gfx1250, wave32).
//   out = atan(gyy * sign(-(gxy+eps)) / (gxx+eps))
// gxx/gxy/gyy are separable 9x9 convs (A9_y⊗D9_x, M9_y⊗M9_x, D9_y⊗A9_x) of the
// raw input == interior-exact composition of gaussian5(reflect)+sobel+sobel.
//
// One wave (32 lanes) per workgroup computes one 16(w)x32(h) output tile:
//  - 40x24 input halo -> LDS: interior tiles via 8x global_load_async_to_lds_b128
//    (contiguous, 16B aligned), border tiles via 30x ..._b32 with reflect idx
//  - row pass:  R(40x16) = In(40x24) x Band(24x16)  3 row-blocks {0,16,24}
//  - col pass:  G(16x16) = Band(16x24) x R(24x16)   2 output row-blocks
//  - band fragments: 18x global_load_b64 off one base into __constant__ table
// ---------------------------------------------------------------------------

typedef __attribute__((ext_vector_type(2))) float v2f;
typedef __attribute__((ext_vector_type(8))) float v8f;
typedef __attribute__((ext_vector_type(4))) int v4i;

#define HHH 1024
#define WWW 1024

// Composite 9-tap filters (g = gauss(5, sigma=2), s=[1,2,1]/4, d=[-1,0,1]/2),
// zero-padded so that index d = (kk - lo) + 15 in [0,37] needs no range check.
//   row 0: A9 = g*s*s   row 1: D9 = g*d*d   row 2: M9 = g*s*d   (taps at +15)
__device__ __constant__ static const float CFPAD[3][40] = {
    {0, 0, 0, 0, 0, 0, 0, 0, 0, 0, 0, 0, 0, 0, 0,
     0.009529321f,  0.051982366f,  0.128347448f,  0.198017634f,  0.224246462f,
     0.198017634f,  0.128347448f,  0.051982366f,  0.009529321f,
     0, 0, 0, 0, 0, 0, 0, 0, 0, 0, 0, 0, 0, 0, 0, 0},
    {0, 0, 0, 0, 0, 0, 0, 0, 0, 0, 0, 0, 0, 0, 0,
     0.038117285f,  0.055460325f, -0.013389790f, -0.055460325f, -0.049454990f,
    -0.055460325f, -0.013389790f,  0.055460325f,  0.038117285f,
     0, 0, 0, 0, 0, 0, 0, 0, 0, 0, 0, 0, 0, 0, 0, 0},
    {0, 0, 0, 0, 0, 0, 0, 0, 0, 0, 0, 0, 0, 0, 0,
    -0.019058643f, -0.065847448f, -0.086882715f, -0.052457658f,  0.000000000f,
     0.052457658f,  0.086882715f,  0.065847448f,  0.019058643f,
     0, 0, 0, 0, 0, 0, 0, 0, 0, 0, 0, 0, 0, 0, 0, 0}};

#if __has_builtin(__builtin_amdgcn_global_load_async_to_lds_b32) &&  \
    __has_builtin(__builtin_amdgcn_global_load_async_to_lds_b128) && \
    __has_builtin(__builtin_amdgcn_s_wait_asynccnt)
#define USE_ASYNC_LDS 1
#else
#define USE_ASYNC_LDS 0
#endif

typedef __attribute__((address_space(1))) int* gint_p;
typedef __attribute__((address_space(3))) int* lint_p;
typedef __attribute__((address_space(1))) v4i* g4i_p;
typedef __attribute__((address_space(3))) v4i* l4i_p;

__device__ __forceinline__ void g2l_async_b32(const float* g, float* l) {
#if USE_ASYNC_LDS
  __builtin_amdgcn_global_load_async_to_lds_b32(
      (gint_p)(void*)g, (lint_p)(void*)l, /*imm offset*/ 0, /*cpol*/ 0);
#else
  *l = *g;
#endif
}

__device__ __forceinline__ void g2l_async_b128(const float* g, float* l) {
#if USE_ASYNC_LDS
  __builtin_amdgcn_global_load_async_to_lds_b128(
      (g4i_p)(void*)g, (l4i_p)(void*)l, /*imm offset*/ 0, /*cpol*/ 0);
#else
  l[0] = g[0]; l[1] = g[1]; l[2] = g[2]; l[3] = g[3];
#endif
}

__device__ __forceinline__ void wait_async_all() {
#if USE_ASYNC_LDS
  __builtin_amdgcn_s_wait_asynccnt(0);
#endif
}

__device__ __forceinline__ int reflect_i(int v, int n) {
  v = (v < 0) ? -v : v;                // jnp.pad 'reflect' (no edge repeat)
  v = (v >= n) ? (2 * n - 2 - v) : v;
  return v;
}

__device__ __forceinline__ v8f wmma_f32_16x16x4(v2f a, v2f b, v8f c) {
  return __builtin_amdgcn_wmma_f32_16x16x4_f32(false, a, false, b, (short)0, c,
                                               false, false);
}

__global__ __launch_bounds__(32) void normal_angle_kernel(
    const float* __restrict__ in, float* __restrict__ out) {
  __shared__ float inT[40][28];      // 40x24 halo; stride 28 floats = 112B:
                                     // rows 16B aligned, bank-conflict-free
  __shared__ float Rq[3][40][16];    // row-pass results: Rd, Ra, Rm

  const int lane = threadIdx.x;
  const int lo = lane & 15;          // N / M sub-index within half-wave
  const int hi = lane >> 4;          // half-wave select
  const int x0 = blockIdx.x << 4;    // 16-wide tiles
  const int y0 = blockIdx.y << 5;    // 32-tall tiles
  const int b  = blockIdx.z;

  // ---- Stage 0a: kick off async halo copy (40 rows x 24 cols) ------------
  const float* base = in + ((size_t)b << 20);
  const bool interior = (blockIdx.x >= 1) & (blockIdx.x <= 62) &
                        (blockIdx.y >= 1) & (blockIdx.y <= 30);
  if (interior) {
    // Rows y0-4+j contiguous; 16B-aligned chunks of 4 floats, 6 per row.
    const float* rbase = base + ((size_t)(y0 - 4) << 10) + (x0 - 4);
#pragma unroll
    for (int t = 0; t < 8; ++t) {
      int idx = lane + (t << 5);     // 0..255, need 0..239
      if (idx < 240) {
        int j = idx / 6;
        int c = (idx - j * 6) << 2;  // 0,4,...,20
        g2l_async_b128(rbase + ((size_t)j << 10) + c, &inT[j][c]);
      }
    }
  } else {
#pragma unroll 1
    for (int t = 0; t < 30; ++t) {   // 960 elements, rolled (border is rare)
      int idx = lane + (t << 5);
      int j = idx / 24;
      int i = idx - j * 24;
      int gy = reflect_i(y0 - 4 + j, HHH);
      int gx = reflect_i(x0 - 4 + i, WWW);
      g2l_async_b32(base + ((size_t)gy << 10) + gx, &inT[j][i]);
    }
  }

  // ---- Stage 0b: band fragments from constant memory (overlaps async) ----
  // Row pass B-operand: B[K][N] = f[K-N]; col pass A-operand: A[M][K]=f[K-M].
  // Both are v2f{ f[kk-lo], f[kk+1-lo] }, kk = 4*ks + 2*hi. Single per-lane
  // base pointer; f/ks offsets are compile-time immediates.
  v2f bf[3][6];
  {
    const float* q = &CFPAD[0][15 + (hi << 1) - lo];
#pragma unroll
    for (int f = 0; f < 3; ++f) {
#pragma unroll
      for (int ks = 0; ks < 6; ++ks) {
        bf[f][ks].x = q[f * 40 + (ks << 2)];
        bf[f][ks].y = q[f * 40 + (ks << 2) + 1];
      }
    }
  }

  wait_async_all();
  __syncthreads();

  // ---- Stage 1: row pass  R_f(40x16) = In(40x24) x Band_f(24x16) ---------
  // Row blocks {0,16,24}: rows 24..31 recomputed identically (no predicate).
  // x-direction filters per quantity: gxx->D9(1), gyy->A9(0), gxy->M9(2)
  const int XF[3] = {1, 0, 2};
  const int RB[3] = {0, 16, 24};
#pragma unroll
  for (int rbi = 0; rbi < 3; ++rbi) {
    const int r0 = RB[rbi];
    const int m = r0 + lo;           // <= 39, always in range
    v2f a[6];                        // A: data fragments, shared by 3 filters
#pragma unroll
    for (int ks = 0; ks < 6; ++ks) {
      const int kk = (ks << 2) + (hi << 1);
      a[ks].x = inT[m][kk];
      a[ks].y = inT[m][kk + 1];
    }
#pragma unroll
    for (int f = 0; f < 3; ++f) {
      v8f acc = {0.f, 0.f, 0.f, 0.f, 0.f, 0.f, 0.f, 0.f};
#pragma unroll
      for (int ks = 0; ks < 6; ++ks)
        acc = wmma_f32_16x16x4(a[ks], bf[XF[f]][ks], acc);
#pragma unroll
      for (int r = 0; r < 8; ++r) Rq[f][r0 + r + (hi << 3)][lo] = acc[r];
    }
  }
  __syncthreads();

  // ---- Stage 2+3: col pass + angle, per 16-row output block --------------
  // y-direction filters per quantity: gxx->A9(0), gyy->D9(1), gxy->M9(2)
  float* ob_ = out + ((size_t)b << 20);
#pragma unroll
  for (int obk = 0; obk < 2; ++obk) {
    const int rbase = obk << 4;
    v8f g3[3];
#pragma unroll
    for (int f = 0; f < 3; ++f) {
      v8f acc = {0.f, 0.f, 0.f, 0.f, 0.f, 0.f, 0.f, 0.f};
#pragma unroll
      for (int ks = 0; ks < 6; ++ks) {
        const int kk = (ks << 2) + (hi << 1);
        v2f bb;                      // B: data, B[K][N]
        bb.x = Rq[f][rbase + kk][lo];
        bb.y = Rq[f][rbase + kk + 1][lo];
        acc = wmma_f32_16x16x4(bf[f][ks], bb, acc);
      }
      g3[f] = acc;
    }
#pragma unroll
    for (int r = 0; r < 8; ++r) {
      const float xx = g3[0][r] + 1e-6f;
      const float yy = g3[1][r];
      const float v  = g3[2][r] + 1e-6f;
      const float s  = (v > 0.f) ? -1.f : ((v < 0.f) ? 1.f : 0.f);
      const float ang = atanf(__fdividef(yy * s, xx));
      const int row = y0 + (obk << 4) + r + (hi << 3);
      ob_[((size_t)row << 10) + (x0 + lo)] = ang;
    }
  }
}

extern "C" void kernel_launch(void* const* d_in, const int* in_sizes, int n_in,
                              void* d_out, int out_size, void* d_ws,
                              size_t ws_size, hipStream_t stream) {
  (void)in_sizes; (void)n_in; (void)out_size; (void)d_ws; (void)ws_size;
  const float* in = (const float*)d_in[0];
  float* out = (float*)d_out;
  dim3 grid(WWW / 16, HHH / 32, 16);  // 64 x 32 tiles x 16 batch
  dim3 block(32, 1, 1);               // one wave32 per 16x32 tile
  normal_angle_kernel<<<grid, block, 0, stream>>>(in, out);
}